// SOMLabelling_21981642621440
// MI455X (gfx1250) — compile-verified
//
#include <hip/hip_runtime.h>
#include <math.h>

// Problem sizes (fixed by the reference)
#define B_SZ 2048
#define N_SZ 1024
#define D_SZ 128
#define C_SZ 10

typedef __attribute__((ext_vector_type(2))) float v2f;
typedef __attribute__((ext_vector_type(8))) float v8f;

// ---------------------------------------------------------------------------
// Kernel 0: row norms.  rows 0..2047 -> xnorm, rows 2048..3071 -> nnorm.
// One wave32 per row, coalesced strided loads, shfl-xor reduction.
// ---------------------------------------------------------------------------
__global__ void __launch_bounds__(256) norms_kernel(
    const float* __restrict__ x, const float* __restrict__ neurons,
    float* __restrict__ xnorm, float* __restrict__ nnorm) {
  const int wave = threadIdx.x >> 5;
  const int lane = threadIdx.x & 31;
  const int row  = blockIdx.x * 8 + wave;            // 0..3071 (grid = 384)

  const float* src;
  float* dst;
  int r;
  if (row < B_SZ) { src = x;       dst = xnorm; r = row; }
  else            { src = neurons; dst = nnorm; r = row - B_SZ; }

  float s = 0.f;
#pragma unroll
  for (int i = 0; i < D_SZ / 32; ++i) {
    float v = src[r * D_SZ + lane + i * 32];
    s += v * v;
  }
#pragma unroll
  for (int m = 16; m >= 1; m >>= 1) s += __shfl_xor(s, m, 32);
  if (lane == 0) dst[r] = s;
}

// ---------------------------------------------------------------------------
// Kernel 1: init output activities / counts sections (accumulated into later).
// ---------------------------------------------------------------------------
__global__ void init_out_kernel(const float* __restrict__ act_in,
                                const int* __restrict__ cnt_in,
                                float* __restrict__ act_out,
                                int* __restrict__ cnt_out) {
  int i = blockIdx.x * blockDim.x + threadIdx.x;
  if (i < N_SZ * C_SZ) act_out[i] = act_in[i];
  if (i < C_SZ)        cnt_out[i] = cnt_in[i];
}

// ---------------------------------------------------------------------------
// Kernel 2: d2[b,n] = xnorm[b] + nnorm[n] - 2 * (x · neuronsT)[b,n]
// via V_WMMA_F32_16X16X4_F32.  Block = 256 threads = 8 waves.
// Block tile: 16 (b-rows) x 128 (n-cols); one 16x16 n-tile per wave.
// x tile (16x128 f32) staged in LDS with padded stride 132 floats so the
// 16 lanes reading 16 different rows at the same k hit 16 distinct banks.
//
// fp32 16x4 A fragment layout (ISA 7.12.2): lane L holds row (L&15),
// K = 2*(L>>4) + j in VGPR j.  B (4x16) mirrors it with N = (L&15).
// ---------------------------------------------------------------------------
#define XPAD 132

__global__ void __launch_bounds__(256) d2_wmma_kernel(
    const float* __restrict__ x, const float* __restrict__ neurons,
    const float* __restrict__ xnorm, const float* __restrict__ nnorm,
    float* __restrict__ d2) {
  __shared__ float lds_x[16 * XPAD];

  const int tid  = threadIdx.x;
  const int wave = tid >> 5;
  const int lane = tid & 31;
  const int b0   = blockIdx.x * 16;               // grid.x = 128
  const int n0   = blockIdx.y * 128 + wave * 16;  // grid.y = 8

  // Stage 16x128 x-tile into LDS: 512 float4, 256 threads x 2 iters.
  const float4* xg = reinterpret_cast<const float4*>(x);
#pragma unroll
  for (int i = 0; i < 2; ++i) {
    int f   = tid + i * 256;        // float4 index 0..511
    int row = f >> 5;               // 32 float4 per row
    int c4  = f & 31;
    float4 v = xg[(b0 + row) * (D_SZ / 4) + c4];
    *reinterpret_cast<float4*>(&lds_x[row * XPAD + c4 * 4]) = v;
  }
  __syncthreads();

  const int m  = lane & 15;          // row within fragment
  const int kb = (lane >> 4) * 2;    // K sub-pair for this half-wave
  const float* ap = &lds_x[m * XPAD + kb];
  const float* bp = neurons + (size_t)(n0 + m) * D_SZ + kb;

  v8f acc = {};                      // C starts at 0
#pragma unroll
  for (int k0 = 0; k0 < D_SZ; k0 += 4) {
    v2f a = *reinterpret_cast<const v2f*>(ap + k0);   // ds_load_b64
    v2f b = *reinterpret_cast<const v2f*>(bp + k0);   // global_load_b64
    acc = __builtin_amdgcn_wmma_f32_16x16x4_f32(
        /*neg_a=*/false, a, /*neg_b=*/false, b,
        /*c_mod=*/(short)0, acc, /*reuse_a=*/false, /*reuse_b=*/false);
  }

  // Epilogue: C/D layout -> lane L, VGPR r holds (M = r + 8*(L>>4), N = L&15)
  const int ncol = n0 + (lane & 15);
  const float nn = nnorm[ncol];
#pragma unroll
  for (int r = 0; r < 8; ++r) {
    int brow = b0 + r + 8 * (lane >> 4);
    float v  = xnorm[brow] + nn - 2.0f * acc[r];
    d2[(size_t)brow * N_SZ + ncol] = fmaxf(v, 0.0f);  // guard fp rounding < 0
  }
}

// ---------------------------------------------------------------------------
// Kernel 3: per batch-row b (one wave32 each):
//   argmin_n d2[b,n]  (first-index tie-break, matches jnp.argmin)
//   g_norm = exp(-sqrt(d2)/sig) / exp(-sqrt(d2_min)/sig)   (0 if gb == 0)
//   activities[n, cls[b]] += g_norm;  counts[cls[b]] += 1
//   out[b,:] = labels[bmu,:]
// ---------------------------------------------------------------------------
__global__ void __launch_bounds__(256) reduce_scatter_kernel(
    const float* __restrict__ d2, const float* __restrict__ y,
    const float* __restrict__ labels, const float* __restrict__ sigma,
    float* __restrict__ out, float* __restrict__ act_out,
    int* __restrict__ cnt_out) {
  const int wave = threadIdx.x >> 5;
  const int lane = threadIdx.x & 31;
  const int b    = blockIdx.x * 8 + wave;          // grid = 256
  const float sig = sigma[0];

  // cls = argmax over 10 entries of y[b,:] (lane 0, then broadcast)
  int cls = 0;
  if (lane == 0) {
    float best = y[b * C_SZ];
    for (int c = 1; c < C_SZ; ++c) {
      float v = y[b * C_SZ + c];
      if (v > best) { best = v; cls = c; }
    }
  }
  cls = __shfl(cls, 0, 32);

  // strided scan of the 1024-entry row; per-lane first-occurrence argmin
  float vals[N_SZ / 32];
  float vmin = 3.402823466e38f;
  int   imin = 0;
#pragma unroll
  for (int i = 0; i < N_SZ / 32; ++i) {
    int n   = lane + i * 32;
    float v = d2[(size_t)b * N_SZ + n];
    vals[i] = v;
    if (v < vmin) { vmin = v; imin = n; }
  }
  // cross-lane min with lowest-index tie-break
#pragma unroll
  for (int m = 16; m >= 1; m >>= 1) {
    float ov = __shfl_xor(vmin, m, 32);
    int   oi = __shfl_xor(imin, m, 32);
    if (ov < vmin || (ov == vmin && oi < imin)) { vmin = ov; imin = oi; }
  }

  const float gb = __expf(-sqrtf(vmin) / sig);
#pragma unroll
  for (int i = 0; i < N_SZ / 32; ++i) {
    int n   = lane + i * 32;
    float g = __expf(-sqrtf(vals[i]) / sig);
    float gn = (gb == 0.f) ? 0.f : (g / gb);
    atomicAdd(&act_out[n * C_SZ + cls], gn);
  }

  if (lane == 0) atomicAdd(&cnt_out[cls], 1);
  if (lane < C_SZ) out[b * C_SZ + lane] = labels[imin * C_SZ + lane];
}

// ---------------------------------------------------------------------------
// Host launcher.  Inputs (setup_inputs order):
//   0:x(2048x128 f32) 1:y(2048x10 f32) 2:neurons(1024x128 f32)
//   3:labels(1024x10 f32) 4:activities(1024x10 f32) 5:labels_count(10 i32)
//   6:sigma(1 f32)
// d_out = [out 2048x10 f32 | activities 1024x10 f32 | counts 10 i32-bits]
// d_ws  = [xnorm 2048 | nnorm 1024 | pad to 4096 | d2 2048x1024] floats (~8.4MB)
// ---------------------------------------------------------------------------
extern "C" void kernel_launch(void* const* d_in, const int* in_sizes, int n_in,
                              void* d_out, int out_size, void* d_ws, size_t ws_size,
                              hipStream_t stream) {
  const float* x       = (const float*)d_in[0];
  const float* y       = (const float*)d_in[1];
  const float* neurons = (const float*)d_in[2];
  const float* labels  = (const float*)d_in[3];
  const float* act_in  = (const float*)d_in[4];
  const int*   cnt_in  = (const int*)d_in[5];
  const float* sigma   = (const float*)d_in[6];

  float* out     = (float*)d_out;                     // B*C
  float* act_out = out + B_SZ * C_SZ;                 // N*C
  int*   cnt_out = (int*)(act_out + N_SZ * C_SZ);     // C

  float* xnorm = (float*)d_ws;
  float* nnorm = xnorm + B_SZ;
  float* d2    = xnorm + 4096;                        // 8 MB region

  norms_kernel<<<(B_SZ + N_SZ) / 8, 256, 0, stream>>>(x, neurons, xnorm, nnorm);
  init_out_kernel<<<(N_SZ * C_SZ + 255) / 256, 256, 0, stream>>>(
      act_in, cnt_in, act_out, cnt_out);

  dim3 grid(B_SZ / 16, N_SZ / 128);
  d2_wmma_kernel<<<grid, 256, 0, stream>>>(x, neurons, xnorm, nnorm, d2);

  reduce_scatter_kernel<<<B_SZ / 8, 256, 0, stream>>>(
      d2, y, labels, sigma, out, act_out, cnt_out);
}